// ETP_80547816669502
// MI455X (gfx1250) — compile-verified
//
#include <hip/hip_runtime.h>
#include <math.h>

// ---------------------------------------------------------------------------
// FASTopic ETP Sinkhorn on MI455X (gfx1250, wave32)
//   x: [256, 384] f32   y: [32768, 384] f32
//   out: [1 + 256*32768] f32 = { loss_ETP, transp (row-major) }
// Cost matrix GEMM uses V_WMMA_F32_16X16X4_F32 (fp32-exact, matches reference
// precision). log_K0 (32 MB) lives in workspace and stays L2-resident
// (192 MB L2) across the ~100 Sinkhorn sweeps.
// ---------------------------------------------------------------------------

typedef float v2f __attribute__((ext_vector_type(2)));
typedef float v8f __attribute__((ext_vector_type(8)));

#define N_TOPIC 256
#define N_WORD  32768
#define DIM     384
#define ALPHA   20.0f
#define STOP_THR 0.005f
#define LOG_A  (-5.545177444479562f)    // log(1/256)
#define LOG_B  (-10.397207708399180f)   // log(1/32768)
#define NEG_BIG (-3.402823466e+38f)

// ---- online logsumexp helpers ---------------------------------------------
__device__ __forceinline__ void lse_accum(float& m, float& s, float v) {
  if (v > m) { s = s * __expf(m - v) + 1.0f; m = v; }
  else       { s += __expf(v - m); }
}
__device__ __forceinline__ void lse_merge(float& m, float& s, float m2, float s2) {
  if (m2 > m) { s = s * __expf(m - m2) + s2; m = m2; }
  else        { s += s2 * __expf(m2 - m); }
}

// ---- init all carried state (must run every launch: no stale state) -------
__global__ void init_state(float* cum_u, float* log_u, float* cum_v,
                           float* log_v, int* done) {
  int idx = blockIdx.x * 256 + threadIdx.x;
  if (idx < N_WORD)  { cum_v[idx] = 0.f; log_v[idx] = 0.f; }
  if (idx < N_TOPIC) { cum_u[idx] = 0.f; log_u[idx] = 0.f; }
  if (idx == 0) *done = 0;
}

// ---- row squared norms: one block per row ---------------------------------
__global__ __launch_bounds__(128) void sqnorm_kernel(const float* __restrict__ v,
                                                     float* __restrict__ out) {
  __shared__ float red[128];
  int r = blockIdx.x, t = threadIdx.x;
  float s = 0.f;
  for (int c = t; c < DIM; c += 128) { float a = v[(size_t)r * DIM + c]; s += a * a; }
  red[t] = s; __syncthreads();
  for (int off = 64; off > 0; off >>= 1) {
    if (t < off) red[t] += red[t + off];
    __syncthreads();
  }
  if (t == 0) out[r] = red[0];
}

// ---- log_K0 = -alpha*(xx_i + yy_j - 2*x.yT) via fp32 WMMA -----------------
// One wave (32 threads) per 16x16 output tile; K loop = 96 x WMMA 16x16x4 f32.
// A layout: lane l: M = l&15, K = vgpr + 2*(l>>4)  -> contiguous f32 pair.
// C layout: vgpr r, lane l: M = r + 8*(l>>4), N = l&15.
__global__ __launch_bounds__(32) void gemm_logk0(const float* __restrict__ x,
                                                 const float* __restrict__ y,
                                                 const float* __restrict__ xx,
                                                 const float* __restrict__ yy,
                                                 float* __restrict__ logK0) {
  const int tileN = blockIdx.x << 4;   // gridDim.x = 2048
  const int tileM = blockIdx.y << 4;   // gridDim.y = 16
  const int lane  = threadIdx.x;
  const int half  = lane >> 4;
  const int l16   = lane & 15;

  const float* xrow = x + (size_t)(tileM + l16) * DIM;
  const float* ycol = y + (size_t)(tileN + l16) * DIM;

  v8f acc = {};
  #pragma unroll 4
  for (int k = 0; k < DIM; k += 4) {
    const int kk = k + 2 * half;
    v2f a = *(const v2f*)(xrow + kk);
    v2f b = *(const v2f*)(ycol + kk);
    acc = __builtin_amdgcn_wmma_f32_16x16x4_f32(
        /*neg_a=*/false, a, /*neg_b=*/false, b,
        /*c_mod=*/(short)0, acc, /*reuse_a=*/false, /*reuse_b=*/false);
  }

  const int   col = tileN + l16;
  const float yyc = yy[col];
  #pragma unroll
  for (int r = 0; r < 8; ++r) {
    int row = tileM + r + 8 * half;
    // -alpha*(xx+yy-2*dot) = 2*alpha*dot - alpha*xx - alpha*yy
    float lk0 = 2.0f * ALPHA * acc[r] - ALPHA * xx[row] - ALPHA * yyc;
    logK0[(size_t)row * N_WORD + col] = lk0;
  }
}

// ---- column sweep: log_v[j] = log_b - cum_v[j] - LSE_i(logK0[i,j]+U[i]) ---
__global__ __launch_bounds__(256) void col_lse_kernel(const float* __restrict__ logK0,
    const float* __restrict__ cum_u, const float* __restrict__ log_u,
    const float* __restrict__ cum_v, float* __restrict__ log_v,
    const int* __restrict__ done) {
  if (*done) return;
  __shared__ float U[N_TOPIC];
  const int t = threadIdx.x;
  U[t] = cum_u[t] + log_u[t];
  __syncthreads();
  const int j = blockIdx.x * 256 + t;          // 128 blocks cover 32768 cols
  float m = NEG_BIG, s = 0.f;
  for (int i = 0; i < N_TOPIC; ++i)
    lse_accum(m, s, logK0[(size_t)i * N_WORD + j] + U[i]);
  log_v[j] = LOG_B - cum_v[j] - (m + __logf(s));
}

// ---- row sweep: log_u[i] = log_a - cum_u[i] - LSE_j(logK0[i,j]+cum_v+log_v)
__global__ __launch_bounds__(256) void row_lse_kernel(const float* __restrict__ logK0,
    const float* __restrict__ cum_v, const float* __restrict__ log_v,
    const float* __restrict__ cum_u, float* __restrict__ log_u,
    const int* __restrict__ done) {
  if (*done) return;
  __shared__ float sm[256], ss[256];
  const int i = blockIdx.x;                    // one block per row
  const int t = threadIdx.x;
  const float* rowp = logK0 + (size_t)i * N_WORD;
  float m = NEG_BIG, s = 0.f;
  for (int j = t; j < N_WORD; j += 256)
    lse_accum(m, s, rowp[j] + cum_v[j] + log_v[j]);
  sm[t] = m; ss[t] = s; __syncthreads();
  for (int off = 128; off > 0; off >>= 1) {
    if (t < off) lse_merge(sm[t], ss[t], sm[t + off], ss[t + off]);
    __syncthreads();
  }
  if (t == 0) log_u[i] = LOG_A - cum_u[i] - (sm[0] + __logf(ss[0]));
}

// ---- absorption: cum += log; log = 0 --------------------------------------
__global__ void absorb_kernel(float* cum, float* logv, int n, const int* done) {
  if (*done) return;
  int idx = blockIdx.x * 256 + threadIdx.x;
  if (idx < n) { cum[idx] += logv[idx]; logv[idx] = 0.f; }
}

// ---- marginal error, column part: per-block max of |col_j - 1/m| ----------
__global__ __launch_bounds__(256) void err_col_kernel(const float* __restrict__ logK0,
    const float* __restrict__ cum_u, const float* __restrict__ cum_v,
    float* __restrict__ part, const int* __restrict__ done) {
  if (*done) return;
  __shared__ float U[N_TOPIC];
  __shared__ float red[256];
  const int t = threadIdx.x;
  U[t] = cum_u[t];
  __syncthreads();
  const int j = blockIdx.x * 256 + t;
  float m = NEG_BIG, s = 0.f;
  for (int i = 0; i < N_TOPIC; ++i)
    lse_accum(m, s, logK0[(size_t)i * N_WORD + j] + U[i]);
  float colv = __expf(m + __logf(s) + cum_v[j]);
  red[t] = fabsf(colv - 1.0f / N_WORD);
  __syncthreads();
  for (int off = 128; off > 0; off >>= 1) {
    if (t < off) red[t] = fmaxf(red[t], red[t + off]);
    __syncthreads();
  }
  if (t == 0) part[blockIdx.x] = red[0];
}

// ---- marginal error, row part: |row_i - 1/n| per block (row) --------------
__global__ __launch_bounds__(256) void err_row_kernel(const float* __restrict__ logK0,
    const float* __restrict__ cum_u, const float* __restrict__ cum_v,
    float* __restrict__ part, const int* __restrict__ done) {
  if (*done) return;
  __shared__ float sm[256], ss[256];
  const int i = blockIdx.x;
  const int t = threadIdx.x;
  const float* rowp = logK0 + (size_t)i * N_WORD;
  float m = NEG_BIG, s = 0.f;
  for (int j = t; j < N_WORD; j += 256)
    lse_accum(m, s, rowp[j] + cum_v[j]);
  sm[t] = m; ss[t] = s; __syncthreads();
  for (int off = 128; off > 0; off >>= 1) {
    if (t < off) lse_merge(sm[t], ss[t], sm[t + off], ss[t + off]);
    __syncthreads();
  }
  if (t == 0) {
    float rowv = __expf(sm[0] + __logf(ss[0]) + cum_u[i]);
    part[i] = fabsf(rowv - 1.0f / N_TOPIC);
  }
}

// ---- finalize err: max over 384 partials; set done flag -------------------
__global__ __launch_bounds__(256) void err_finalize(const float* __restrict__ part,
                                                    int* done) {
  if (*done) return;
  __shared__ float red[256];
  const int t = threadIdx.x;
  float m = part[t];                               // [0..255]: 128 col + row head
  if (t < 128) m = fmaxf(m, part[256 + t]);        // [256..383]
  red[t] = m; __syncthreads();
  for (int off = 128; off > 0; off >>= 1) {
    if (t < off) red[t] = fmaxf(red[t], red[t + off]);
    __syncthreads();
  }
  if (t == 0) *done = (red[0] <= STOP_THR) ? 1 : 0;
}

// ---- transp = exp(logK0 + U_i + V_j); loss partials -----------------------
// 2048 contiguous elems per block -> row index i constant per block.
__global__ __launch_bounds__(256) void transp_loss(const float* __restrict__ logK0,
    const float* __restrict__ cum_u, const float* __restrict__ log_u,
    const float* __restrict__ cum_v, const float* __restrict__ log_v,
    float* __restrict__ transp, float* __restrict__ losspart) {
  __shared__ float red[256];
  const int t  = threadIdx.x;
  const int i  = blockIdx.x >> 4;                  // 16 blocks per row
  const int j0 = (blockIdx.x & 15) << 11;          // *2048
  const float Ui = cum_u[i] + log_u[i];
  const float* rowK = logK0 + (size_t)i * N_WORD;
  float* rowT = transp + (size_t)i * N_WORD;
  float acc = 0.f;
  #pragma unroll
  for (int e = 0; e < 8; ++e) {
    int j = j0 + e * 256 + t;
    float lk0 = rowK[j];
    float tr  = __expf(lk0 + Ui + cum_v[j] + log_v[j]);
    rowT[j] = tr;
    acc += tr * (lk0 * (-1.0f / ALPHA));           // M = -logK0/alpha
  }
  red[t] = acc; __syncthreads();
  for (int off = 128; off > 0; off >>= 1) {
    if (t < off) red[t] += red[t + off];
    __syncthreads();
  }
  if (t == 0) losspart[blockIdx.x] = red[0];
}

__global__ __launch_bounds__(256) void loss_finalize(const float* __restrict__ part,
                                                     float* __restrict__ out) {
  __shared__ float red[256];
  const int t = threadIdx.x;
  float s = 0.f;
  for (int k = t; k < 4096; k += 256) s += part[k];
  red[t] = s; __syncthreads();
  for (int off = 128; off > 0; off >>= 1) {
    if (t < off) red[t] += red[t + off];
    __syncthreads();
  }
  if (t == 0) out[0] = red[0];
}

// ---------------------------------------------------------------------------
extern "C" void kernel_launch(void* const* d_in, const int* in_sizes, int n_in,
                              void* d_out, int out_size, void* d_ws, size_t ws_size,
                              hipStream_t stream) {
  (void)in_sizes; (void)n_in; (void)out_size; (void)ws_size;
  const float* x = (const float*)d_in[0];   // [256, 384]
  const float* y = (const float*)d_in[1];   // [32768, 384]
  float* out = (float*)d_out;               // [1 + 256*32768]

  // workspace carve-up (~33.8 MB total)
  float* w       = (float*)d_ws;
  float* logK0   = w;                                 // 8,388,608
  float* xx      = logK0 + (size_t)N_TOPIC * N_WORD;  // 256
  float* yy      = xx + N_TOPIC;                      // 32768
  float* cum_u   = yy + N_WORD;                       // 256
  float* log_u   = cum_u + N_TOPIC;                   // 256
  float* cum_v   = log_u + N_TOPIC;                   // 32768
  float* log_v   = cum_v + N_WORD;                    // 32768
  float* errpart = log_v + N_WORD;                    // 384 (128 col + 256 row)
  float* losspart= errpart + 384;                     // 4096
  int*   done    = (int*)(losspart + 4096);           // 1

  // fresh state each launch (graph-replay safe, deterministic)
  init_state<<<128, 256, 0, stream>>>(cum_u, log_u, cum_v, log_v, done);
  sqnorm_kernel<<<N_TOPIC, 128, 0, stream>>>(x, xx);
  sqnorm_kernel<<<N_WORD, 128, 0, stream>>>(y, yy);
  gemm_logk0<<<dim3(N_WORD / 16, N_TOPIC / 16), 32, 0, stream>>>(x, y, xx, yy, logK0);

  // Sinkhorn: fixed 100-iteration schedule; `done` flag reproduces
  // while(err > STOP_THR) semantics (err checked at cpt=1 and cpt=51).
  for (int cpt = 1; cpt <= 100; ++cpt) {
    col_lse_kernel<<<N_WORD / 256, 256, 0, stream>>>(logK0, cum_u, log_u, cum_v,
                                                     log_v, done);
    row_lse_kernel<<<N_TOPIC, 256, 0, stream>>>(logK0, cum_v, log_v, cum_u,
                                                log_u, done);
    if (cpt == 1 || cpt == 51) {
      absorb_kernel<<<1, 256, 0, stream>>>(cum_u, log_u, N_TOPIC, done);
      absorb_kernel<<<N_WORD / 256, 256, 0, stream>>>(cum_v, log_v, N_WORD, done);
      err_col_kernel<<<N_WORD / 256, 256, 0, stream>>>(logK0, cum_u, cum_v,
                                                       errpart, done);
      err_row_kernel<<<N_TOPIC, 256, 0, stream>>>(logK0, cum_u, cum_v,
                                                  errpart + 128, done);
      err_finalize<<<1, 256, 0, stream>>>(errpart, done);
    }
  }

  // transp + loss
  transp_loss<<<4096, 256, 0, stream>>>(logK0, cum_u, log_u, cum_v, log_v,
                                        out + 1, losspart);
  loss_finalize<<<1, 256, 0, stream>>>(losspart, out);
}